// GenAttn_69114613730221
// MI455X (gfx1250) — compile-verified
//
#include <hip/hip_runtime.h>
#include <math.h>

#define HID     2048
#define TWOH    4096
#define SEQL    8192

#define KSPLIT1 8                    // kernel-1 h-split
#define HCHUNK  (HID / KSPLIT1)      // 256
#define KSPLIT2 4                    // kernel-2 K-split
#define KCHUNK  (TWOH / KSPLIT2)     // 1024

typedef __attribute__((ext_vector_type(2))) float v2f;
typedef __attribute__((ext_vector_type(8))) float v8f;

// -------- Kernel 1: pv[hc][e] = sum_{h in chunk hc} hidden[h] * W[h, e]
// Grid (16, 8) x 256 threads = 1024 waves streaming W (32 MB) coalesced.
// Deterministic split-reduction partials (no float atomics).
__global__ __launch_bounds__(256) void k_gemv_v_part(const float* __restrict__ W,
                                                     const float* __restrict__ hidden,
                                                     float* __restrict__ pv) {
    __shared__ float sh[HCHUNK];
    const int hc = blockIdx.y;
    const int h0 = hc * HCHUNK;
    if (threadIdx.x < HCHUNK) sh[threadIdx.x] = hidden[h0 + threadIdx.x];
    __syncthreads();

    const int e = blockIdx.x * 256 + threadIdx.x;
    const float* col = W + (size_t)h0 * TWOH + e;
    float acc = 0.0f;
#pragma unroll 8
    for (int h = 0; h < HCHUNK; ++h)
        acc = fmaf(sh[h], col[(size_t)h * TWOH], acc);
    pv[hc * TWOH + e] = acc;
}

// -------- Kernel 2: spart[ks][s] = dot(enc[s, k0:k0+1024], v[k0:k0+1024])
// via V_WMMA_F32_16X16X4_F32. Grid (64, 4) x 256 = 2048 waves on the dominant
// 128 MB stream; K-loop unrolled 8x so each wave keeps ~2 KB of b64 loads in
// flight per clause. A tile uses the ISA 16x4 f32 layout (lane m<16: K=0,1;
// lanes 16-31: K=2,3). B is the v K-chunk replicated across all 16 columns
// (broadcast from LDS), so every D column accumulates the same row-dot.
// The v-stage also folds the kernel-1 partial reduction (deterministic).
__global__ __launch_bounds__(256) void k_scores_wmma(const float* __restrict__ enc,
                                                     const float* __restrict__ pv,
                                                     float* __restrict__ spart) {
    __shared__ float sv[KCHUNK];
    const int ks = blockIdx.y;           // 0..3
    const int k0 = ks * KCHUNK;
    for (int i = threadIdx.x; i < KCHUNK; i += 256) {
        float s = 0.0f;
#pragma unroll
        for (int c = 0; c < KSPLIT1; ++c) s += pv[c * TWOH + k0 + i];
        sv[i] = s;
    }
    __syncthreads();

    const int wave    = threadIdx.x >> 5;    // 0..7
    const int lane    = threadIdx.x & 31;
    const int laneLo  = lane & 15;
    const int grp     = lane >> 4;           // 0: K=0..1, 1: K=2..3
    const int rowBase = blockIdx.x * 128 + wave * 16;

    const float* rowPtr = enc + (size_t)(rowBase + laneLo) * TWOH + k0 + 2 * grp;

    v8f acc = {};
#pragma unroll 8
    for (int kc = 0; kc < KCHUNK; kc += 4) {
        v2f a = *(const v2f*)(rowPtr + kc);              // A[m, kc + 2g .. +1]
        v2f b;
        b.x = sv[kc + 2 * grp];                          // B[k, n] = v[k0+kc+k] (all n)
        b.y = sv[kc + 2 * grp + 1];
        acc = __builtin_amdgcn_wmma_f32_16x16x4_f32(
            /*neg_a=*/false, a, /*neg_b=*/false, b,
            /*c_mod=*/(short)0, acc, /*reuse_a=*/false, /*reuse_b=*/false);
    }

    // D layout: VGPR i holds M=i (lanes 0-15) and M=8+i (lanes 16-31);
    // all N columns identical -> lane 0 writes M=0..7, lane 16 writes M=8..15.
    if (laneLo == 0) {
#pragma unroll
        for (int i = 0; i < 8; ++i)
            spart[ks * SEQL + rowBase + grp * 8 + i] = acc[i];
    }
}

// -------- Kernel 3: softmax over 8192 scores (sum the 4 K-partials on load).
__global__ __launch_bounds__(1024) void k_softmax(const float* __restrict__ spart,
                                                  float* __restrict__ out) {
    __shared__ float red[32];
    const int tid = threadIdx.x;

    float vals[8];
    float m = -INFINITY;
#pragma unroll
    for (int i = 0; i < 8; ++i) {
        const int idx = tid + i * 1024;
        float s = 0.0f;
#pragma unroll
        for (int ks = 0; ks < KSPLIT2; ++ks) s += spart[ks * SEQL + idx];
        vals[i] = s;
        m = fmaxf(m, s);
    }
    for (int off = 16; off > 0; off >>= 1) m = fmaxf(m, __shfl_xor(m, off));
    if ((tid & 31) == 0) red[tid >> 5] = m;
    __syncthreads();
    if (tid < 32) {
        float t = red[tid];
        for (int off = 16; off > 0; off >>= 1) t = fmaxf(t, __shfl_xor(t, off));
        red[tid] = t;
    }
    __syncthreads();
    m = red[0];

    float s = 0.0f;
#pragma unroll
    for (int i = 0; i < 8; ++i) {
        vals[i] = __expf(vals[i] - m);
        s += vals[i];
    }
    for (int off = 16; off > 0; off >>= 1) s += __shfl_xor(s, off);
    __syncthreads();                       // everyone done reading red[0]
    if ((tid & 31) == 0) red[tid >> 5] = s;
    __syncthreads();
    if (tid < 32) {
        float t = red[tid];
        for (int off = 16; off > 0; off >>= 1) t += __shfl_xor(t, off);
        red[tid] = t;
    }
    __syncthreads();
    const float inv = 1.0f / red[0];
#pragma unroll
    for (int i = 0; i < 8; ++i) out[tid + i * 1024] = vals[i] * inv;
}

extern "C" void kernel_launch(void* const* d_in, const int* in_sizes, int n_in,
                              void* d_out, int out_size, void* d_ws, size_t ws_size,
                              hipStream_t stream) {
    (void)in_sizes; (void)n_in; (void)out_size; (void)ws_size;
    const float* hidden = (const float*)d_in[0];   // [2048]
    const float* enc    = (const float*)d_in[1];   // [8192, 4096]
    const float* W      = (const float*)d_in[2];   // [2048, 4096]
    // d_in[3] (bias) is provably irrelevant: softmax is shift-invariant.

    float* pv    = (float*)d_ws;                   // [8][4096] partial v
    float* spart = pv + KSPLIT1 * TWOH;            // [4][8192] partial scores
    float* out   = (float*)d_out;                  // [8192] f32

    k_gemv_v_part<<<dim3(TWOH / 256, KSPLIT1), 256, 0, stream>>>(W, hidden, pv);
    k_scores_wmma<<<dim3(SEQL / 128, KSPLIT2), 256, 0, stream>>>(enc, pv, spart);
    k_softmax    <<<1, 1024, 0, stream>>>(spart, out);
}